// DGLGCN_21002390077613
// MI455X (gfx1250) — compile-verified
//
#include <hip/hip_runtime.h>
#include <hip/hip_bf16.h>

typedef float v2f __attribute__((ext_vector_type(2)));
typedef float v8f __attribute__((ext_vector_type(8)));

// ---------------- degree / norm kernels ----------------

__global__ void gcn_degrees(const int* __restrict__ src, const int* __restrict__ dst,
                            float* __restrict__ deg_out, float* __restrict__ deg_in, int E)
{
    int e = blockIdx.x * blockDim.x + threadIdx.x;
    if (e < E) {
        atomicAdd(&deg_out[src[e]], 1.0f);
        atomicAdd(&deg_in[dst[e]], 1.0f);
    }
}

__global__ void gcn_norms(const float* __restrict__ deg_out, const float* __restrict__ deg_in,
                          float* __restrict__ nsrc, float* __restrict__ ndst, int N)
{
    int i = blockIdx.x * blockDim.x + threadIdx.x;
    if (i < N) {
        nsrc[i] = rsqrtf(fmaxf(deg_out[i], 1.0f));
        ndst[i] = rsqrtf(fmaxf(deg_in[i], 1.0f));
    }
}

// ---------------- fp32 WMMA GEMM:  out = (X * norm[:,None]) @ W ----------------
// One wave computes a 32x64 strip using V_WMMA_F32_16X16X4_F32.
// Two 16-row A fragments share the same four B fragments each K-step:
// 12 loaded floats per 8 WMMAs (vs 10 per 4 for a 16x64 tile).
// A fragment (16x4 f32): lane L holds A[L%16][2*(L/16)+v], v=0,1
// B fragment (4x16 f32): lane L holds B[2*(L/16)+v][L%16]
// D (16x16 f32, 8 VGPRs): lane L, reg r holds D[r + 8*(L/16)][L%16]
__global__ __launch_bounds__(32) void gcn_gemm_wmma(
    const float* __restrict__ X, const float* __restrict__ norm,
    const float* __restrict__ W, float* __restrict__ out,
    int M, int K, int Ncols)
{
    const int m0   = blockIdx.x * 32;
    const int n0   = blockIdx.y * 64;
    const int lane = threadIdx.x;
    const int row  = lane & 15;
    const int half = lane >> 4;

    int mU = m0 + row;                      // upper 16-row tile
    int mL = m0 + 16 + row;                 // lower 16-row tile
    if (mU >= M) mU = M - 1;                // data clamp only; EXEC stays all-ones
    if (mL >= M) mL = M - 1;
    const float nrmU = norm[mU];
    const float nrmL = norm[mL];
    const float* __restrict__ xrowU = X + (size_t)mU * K;
    const float* __restrict__ xrowL = X + (size_t)mL * K;

    v8f aU0 = {}, aU1 = {}, aU2 = {}, aU3 = {};
    v8f aL0 = {}, aL1 = {}, aL2 = {}, aL3 = {};

    for (int k0 = 0; k0 < K; k0 += 4) {
        const int kb = k0 + half * 2;
        v2f au, al;
        au.x = xrowU[kb]     * nrmU;
        au.y = xrowU[kb + 1] * nrmU;
        al.x = xrowL[kb]     * nrmL;
        al.y = xrowL[kb + 1] * nrmL;

        const float* __restrict__ w0 = W + (size_t)kb * Ncols + n0 + row;
        const float* __restrict__ w1 = w0 + Ncols;
        v2f b0; b0.x = w0[0];  b0.y = w1[0];
        v2f b1; b1.x = w0[16]; b1.y = w1[16];
        v2f b2; b2.x = w0[32]; b2.y = w1[32];
        v2f b3; b3.x = w0[48]; b3.y = w1[48];

        aU0 = __builtin_amdgcn_wmma_f32_16x16x4_f32(false, au, false, b0, (short)0, aU0, false, false);
        aU1 = __builtin_amdgcn_wmma_f32_16x16x4_f32(false, au, false, b1, (short)0, aU1, false, false);
        aU2 = __builtin_amdgcn_wmma_f32_16x16x4_f32(false, au, false, b2, (short)0, aU2, false, false);
        aU3 = __builtin_amdgcn_wmma_f32_16x16x4_f32(false, au, false, b3, (short)0, aU3, false, false);
        aL0 = __builtin_amdgcn_wmma_f32_16x16x4_f32(false, al, false, b0, (short)0, aL0, false, false);
        aL1 = __builtin_amdgcn_wmma_f32_16x16x4_f32(false, al, false, b1, (short)0, aL1, false, false);
        aL2 = __builtin_amdgcn_wmma_f32_16x16x4_f32(false, al, false, b2, (short)0, aL2, false, false);
        aL3 = __builtin_amdgcn_wmma_f32_16x16x4_f32(false, al, false, b3, (short)0, aL3, false, false);
    }

    #pragma unroll
    for (int r = 0; r < 8; ++r) {
        const int mrU = m0 + r + 8 * half;
        if (mrU < M) {
            float* o = out + (size_t)mrU * Ncols + n0 + row;
            o[0]  = aU0[r];
            o[16] = aU1[r];
            o[32] = aU2[r];
            o[48] = aU3[r];
        }
        const int mrL = m0 + 16 + r + 8 * half;
        if (mrL < M) {
            float* o = out + (size_t)mrL * Ncols + n0 + row;
            o[0]  = aL0[r];
            o[16] = aL1[r];
            o[32] = aL2[r];
            o[48] = aL3[r];
        }
    }
}

// ---------------- edge scatter-add:  agg[dst[e]] += msg[src[e]]  ----------------
// One thread per (edge, 4-float chunk); consecutive threads cover consecutive
// chunks of one edge row (coalesced). Node tables fit in 192MB L2, so both
// the gathers and the float atomics are L2-resident.
__global__ void gcn_scatter_add(const float* __restrict__ msg,
                                const int* __restrict__ src, const int* __restrict__ dst,
                                float* __restrict__ agg, int E, int F)
{
    const int chunks = F >> 2;
    int tid = blockIdx.x * blockDim.x + threadIdx.x;
    if (tid >= E * chunks) return;
    int e = tid / chunks;
    int c = (tid - e * chunks) << 2;
    int s = src[e], d = dst[e];
    const float4 v = *reinterpret_cast<const float4*>(msg + (size_t)s * F + c);
    float* p = agg + (size_t)d * F + c;
    atomicAdd(p + 0, v.x);
    atomicAdd(p + 1, v.y);
    atomicAdd(p + 2, v.z);
    atomicAdd(p + 3, v.w);
}

// ---------------- epilogue:  out = (agg * norm_dst[:,None] + b) [relu] ----------------
__global__ void gcn_finalize(const float* __restrict__ agg, const float* __restrict__ ndst,
                             const float* __restrict__ bias, float* __restrict__ out,
                             int N, int F, int do_relu)
{
    int tid = blockIdx.x * blockDim.x + threadIdx.x;
    if (tid >= N * F) return;
    int n = tid / F;
    int f = tid - n * F;
    float v = fmaf(agg[tid], ndst[n], bias[f]);
    if (do_relu) v = fmaxf(v, 0.0f);
    out[tid] = v;
}

// ---------------- launcher ----------------

extern "C" void kernel_launch(void* const* d_in, const int* in_sizes, int n_in,
                              void* d_out, int out_size, void* d_ws, size_t ws_size,
                              hipStream_t stream) {
    const float* x  = (const float*)d_in[0];
    const float* W1 = (const float*)d_in[1];
    const float* b1 = (const float*)d_in[2];
    const float* W2 = (const float*)d_in[3];
    const float* b2 = (const float*)d_in[4];
    const int*   src = (const int*)d_in[5];
    const int*   dst = (const int*)d_in[6];

    const int hidden = in_sizes[2];               // 256
    const int outdim = in_sizes[4];               // 128
    const int indim  = in_sizes[1] / hidden;      // 512
    const int N      = in_sizes[0] / indim;       // 50000
    const int E      = in_sizes[5];               // 800000

    float* ws       = (float*)d_ws;
    float* deg_out  = ws;  ws += N;
    float* deg_in   = ws;  ws += N;
    float* nsrc     = ws;  ws += N;
    float* ndst     = ws;  ws += N;
    float* bufA     = ws;  ws += (size_t)N * hidden;  // h_pre, then h
    float* bufB     = ws;                             // agg1, then h2_pre | agg2

    // degrees and norms (re-zeroed every call: deterministic)
    hipMemsetAsync(deg_out, 0, sizeof(float) * 2 * (size_t)N, stream);
    gcn_degrees<<<(E + 255) / 256, 256, 0, stream>>>(src, dst, deg_out, deg_in, E);
    gcn_norms<<<(N + 255) / 256, 256, 0, stream>>>(deg_out, deg_in, nsrc, ndst, N);

    // ---- layer 1 ----
    dim3 g1((N + 31) / 32, hidden / 64);
    gcn_gemm_wmma<<<g1, 32, 0, stream>>>(x, nsrc, W1, bufA, N, indim, hidden);
    hipMemsetAsync(bufB, 0, sizeof(float) * (size_t)N * hidden, stream);
    {
        int total = E * (hidden / 4);
        gcn_scatter_add<<<(total + 255) / 256, 256, 0, stream>>>(bufA, src, dst, bufB, E, hidden);
    }
    gcn_finalize<<<(N * hidden + 255) / 256, 256, 0, stream>>>(bufB, ndst, b1, bufA, N, hidden, 1);

    // ---- layer 2 ----
    float* h2   = bufB;                         // agg1 fully consumed above
    float* agg2 = bufB + (size_t)N * outdim;
    dim3 g2((N + 31) / 32, outdim / 64);
    gcn_gemm_wmma<<<g2, 32, 0, stream>>>(bufA, nsrc, W2, h2, N, hidden, outdim);
    hipMemsetAsync(agg2, 0, sizeof(float) * (size_t)N * outdim, stream);
    {
        int total = E * (outdim / 4);
        gcn_scatter_add<<<(total + 255) / 256, 256, 0, stream>>>(h2, src, dst, agg2, E, outdim);
    }
    gcn_finalize<<<(N * outdim + 255) / 256, 256, 0, stream>>>(agg2, ndst, b2, (float*)d_out, N, outdim, 0);
}